// MultiPBceloss_10823317586697
// MI455X (gfx1250) — compile-verified
//
#include <hip/hip_runtime.h>
#include <stdint.h>

// Problem constants (match reference)
#define Bt      8
#define Hh      512
#define Ww      512
#define HWc     (Hh * Ww)        // 262144 = 2^18
#define NPLANES (Bt * 2)         // 16 CCL planes (bg, fg per image)
#define NPIX    (Bt * HWc)       // 2097152 pixels per pred
#define GX      512              // CE blocks in x
#define CET     (GX * 256)       // CE threads per pred = 131072

typedef __attribute__((ext_vector_type(2))) float v2f;
typedef __attribute__((ext_vector_type(8))) float v8f;

#if defined(__has_builtin)
#if __has_builtin(__builtin_amdgcn_wmma_f32_16x16x4_f32)
#define HAVE_WMMA_F32X4 1
#endif
#endif

// ---------------- union-find (Komura-style, atomicMin linking) -------------
__device__ inline int findRoot(int* par, int x) {
  int p = par[x];
  while (p != x) { x = p; p = par[x]; }
  return x;
}

__device__ inline void unite(int* par, int a, int b) {
  bool done = false;
  while (!done) {
    a = findRoot(par, a);
    b = findRoot(par, b);
    if (a < b) {
      int old = atomicMin(&par[b], a);
      done = (old == b);
      b = old;
    } else if (b < a) {
      int old = atomicMin(&par[a], b);
      done = (old == a);
      a = old;
    } else {
      done = true;
    }
  }
}

__device__ inline bool plane_mask(float cam, int m) {
  return m ? (cam > 0.6f) : (cam >= 0.2f);
}

// ---------------- kernels --------------------------------------------------
__global__ void k_zero_scal(unsigned* scal) {
  int t = threadIdx.x;
  if (t < 32) scal[t] = 0u;
}

__global__ void k_init(int* __restrict__ parent, unsigned* __restrict__ count) {
  int idx = blockIdx.x * 256 + threadIdx.x;
  int p   = blockIdx.y * HWc + idx;
  parent[p] = p;
  count[p]  = 0u;
}

__global__ void k_merge(int* __restrict__ parent, const float* __restrict__ cams) {
  int idx = blockIdx.x * 256 + threadIdx.x;
  int pid = blockIdx.y;
  int b = pid >> 1, m = pid & 1;
  const float* cb = cams + (size_t)b * HWc;
  float c = cb[idx];
  if (!plane_mask(c, m)) return;
  int x = idx & (Ww - 1);
  int y = idx >> 9;
  int p = pid * HWc + idx;
  if (x > 0) {
    if (plane_mask(cb[idx - 1], m)) unite(parent, p, p - 1);
  }
  if (y > 0) {
    if (plane_mask(cb[idx - Ww], m)) unite(parent, p, p - Ww);
  }
}

__global__ void k_flatten_count(int* __restrict__ parent, unsigned* __restrict__ count) {
  int idx = blockIdx.x * 256 + threadIdx.x;
  int p   = blockIdx.y * HWc + idx;
  int r = findRoot(parent, p);
  parent[p] = r;
  atomicAdd(&count[r], 1u);  // non-mask pixels are singleton roots; harmless (see keep test)
}

__global__ void k_maxarea(const unsigned* __restrict__ count, unsigned* __restrict__ scal) {
  __shared__ unsigned sm[256];
  int idx = blockIdx.x * 256 + threadIdx.x;
  int pid = blockIdx.y;
  sm[threadIdx.x] = count[pid * HWc + idx];
  __syncthreads();
  for (int s = 128; s > 0; s >>= 1) {
    if (threadIdx.x < s) {
      unsigned o = sm[threadIdx.x + s];
      if (o > sm[threadIdx.x]) sm[threadIdx.x] = o;
    }
    __syncthreads();
  }
  if (threadIdx.x == 0) atomicMax(&scal[pid], sm[0]);
}

__global__ void k_keep(const int* __restrict__ parent, const unsigned* __restrict__ count,
                       const unsigned* __restrict__ scal, const float* __restrict__ cams,
                       unsigned char* __restrict__ keepfg, unsigned char* __restrict__ keepbg,
                       unsigned* __restrict__ nfg, unsigned* __restrict__ nkeepbg) {
  int idx = blockIdx.x * 256 + threadIdx.x;
  int b   = blockIdx.y;
  int g   = b * HWc + idx;
  float c = cams[g];
  bool mb = (c >= 0.2f);
  bool mf = (c > 0.6f);
  int pb = (2 * b) * HWc + idx;
  int pf = (2 * b + 1) * HWc + idx;
  // keep iff 2*area > max_area  (== area > 0.5*max_area)
  bool kb = mb && (2ull * (unsigned long long)count[parent[pb]] > (unsigned long long)scal[2 * b]);
  bool kf = mf && (2ull * (unsigned long long)count[parent[pf]] > (unsigned long long)scal[2 * b + 1]);
  keepbg[g] = kb ? 1 : 0;
  keepfg[g] = kf ? 1 : 0;
  unsigned long long balF = __ballot(kf);
  unsigned long long balB = __ballot(kb);
  if ((threadIdx.x & (warpSize - 1)) == 0) {
    atomicAdd(nfg,     (unsigned)__popcll(balF));
    atomicAdd(nkeepbg, (unsigned)__popcll(balB));
  }
}

// Cross-entropy partial sums. blockIdx.z = pred index i.
// Per wave, 64 NLL contributions are packed into the 16x4 f32 A operand and
// accumulated into a 16x16 f32 C tile via v_wmma_f32_16x16x4_f32 with B = ones
// (C[m][n] += sum_k A[m][k]); the C tile is the running reduction accumulator.
__global__ void k_ce(const float* __restrict__ preds,
                     const unsigned char* __restrict__ keepfg,
                     const unsigned char* __restrict__ keepbg,
                     float* __restrict__ part) {
  const int i = blockIdx.z;
  const int t = blockIdx.x * 256 + threadIdx.x;     // 0..CET-1
  const float* base = preds + (size_t)i * (size_t)Bt * 2u * (size_t)HWc;

#if HAVE_WMMA_F32X4
  v8f cf = {};
  v8f cb = {};
  const v2f ones = {1.0f, 1.0f};
#else
  float accf = 0.0f, accb = 0.0f;
#endif

#pragma unroll
  for (int k = 0; k < 8; ++k) {
    float fv[2], gv[2];
#pragma unroll
    for (int h = 0; h < 2; ++h) {
      int p  = (k * 2 + h) * CET + t;               // global pixel id, covers [0, NPIX)
      int b  = p >> 18;                             // HWc == 2^18
      int hw = p & (HWc - 1);
      const float* q = base + ((size_t)b * 2u) * (size_t)HWc + (size_t)hw;
      if (k < 7) __builtin_prefetch(q + 2u * (size_t)CET, 0, 1);  // global_prefetch_b8
      float x0 = q[0];
      float x1 = q[HWc];
      float mx  = fmaxf(x0, x1);
      float mn  = fminf(x0, x1);
      float lse = mx + log1pf(expf(mn - mx));
      float nll0 = lse - x0;                        // -log p(class 0)
      float nll1 = lse - x1;                        // -log p(class 1)
      fv[h] = keepfg[p] ? nll1 : 0.0f;              // fg target=1, valid where keep_fg
      gv[h] = keepbg[p] ? 0.0f : nll0;              // bg target=0, valid where !keep_bg
    }
#if HAVE_WMMA_F32X4
    v2f af = {fv[0], fv[1]};
    v2f ag = {gv[0], gv[1]};
    cf = __builtin_amdgcn_wmma_f32_16x16x4_f32(false, af, false, ones, (short)0, cf, false, false);
    cb = __builtin_amdgcn_wmma_f32_16x16x4_f32(false, ag, false, ones, (short)0, cb, false, false);
#else
    accf += fv[0] + fv[1];
    accb += gv[0] + gv[1];
#endif
  }

  float tf, tb;
#if HAVE_WMMA_F32X4
  // lanes 0-15 hold rows 0-7 of C, lanes 16-31 rows 8-15; every column identical
  float sf = cf[0] + cf[1] + cf[2] + cf[3] + cf[4] + cf[5] + cf[6] + cf[7];
  float sb = cb[0] + cb[1] + cb[2] + cb[3] + cb[4] + cb[5] + cb[6] + cb[7];
  tf = __shfl(sf, 0, 32) + __shfl(sf, 16, 32);
  tb = __shfl(sb, 0, 32) + __shfl(sb, 16, 32);
#else
  for (int o = 16; o > 0; o >>= 1) {
    accf += __shfl_down(accf, o, 32);
    accb += __shfl_down(accb, o, 32);
  }
  tf = __shfl(accf, 0, 32);
  tb = __shfl(accb, 0, 32);
#endif

  __shared__ float lf[8], lb[8];
  int w = threadIdx.x >> 5;
  if ((threadIdx.x & 31) == 0) { lf[w] = tf; lb[w] = tb; }
  __syncthreads();
  if (threadIdx.x == 0) {
    float af2 = 0.0f, ab2 = 0.0f;
    for (int j = 0; j < 8; ++j) { af2 += lf[j]; ab2 += lb[j]; }
    part[(i * 2 + 0) * GX + blockIdx.x] = af2;
    part[(i * 2 + 1) * GX + blockIdx.x] = ab2;
  }
}

__global__ void k_final(const float* __restrict__ part, const unsigned* __restrict__ scal,
                        float* __restrict__ out) {
  __shared__ float s[6];
  int t = threadIdx.x;
  if (t < 6) {
    float acc = 0.0f;
    for (int j = 0; j < GX; ++j) acc += part[t * GX + j];   // fixed order -> deterministic
    s[t] = acc;
  }
  __syncthreads();
  if (t == 0) {
    unsigned nfg = scal[16];
    unsigned nbg = (unsigned)NPIX - scal[17];
    float dfg = (float)(nfg > 0u ? nfg : 1u);
    float dbg = (float)(nbg > 0u ? nbg : 1u);
    float loss = 0.0f;
    for (int i = 0; i < 3; ++i)
      loss += s[i * 2 + 0] / dfg + s[i * 2 + 1] / dbg;
    out[0] = loss;
  }
}

// ---------------- launch ---------------------------------------------------
extern "C" void kernel_launch(void* const* d_in, const int* in_sizes, int n_in,
                              void* d_out, int out_size, void* d_ws, size_t ws_size,
                              hipStream_t stream) {
  const float* preds = (const float*)d_in[0];  // [3,B,2,H,W]
  const float* cams  = (const float*)d_in[1];  // [B,1,H,W]

  char* ws = (char*)d_ws;
  size_t off = 0;
  int*      parent = (int*)(ws + off);       off += (size_t)NPLANES * HWc * sizeof(int);
  unsigned* count  = (unsigned*)(ws + off);  off += (size_t)NPLANES * HWc * sizeof(unsigned);
  unsigned char* keepfg = (unsigned char*)(ws + off); off += (size_t)NPIX;
  unsigned char* keepbg = (unsigned char*)(ws + off); off += (size_t)NPIX;
  unsigned* scal = (unsigned*)(ws + off);    off += 256;           // [0..15]=max_area, [16]=Nfg, [17]=NkeepBg
  float*    part = (float*)(ws + off);       off += (size_t)3 * 2 * GX * sizeof(float);

  dim3 blk(256);
  dim3 gplane(HWc / 256, NPLANES);
  dim3 gimg(HWc / 256, Bt);

  k_zero_scal<<<1, 32, 0, stream>>>(scal);
  k_init<<<gplane, blk, 0, stream>>>(parent, count);
  k_merge<<<gplane, blk, 0, stream>>>(parent, cams);
  k_flatten_count<<<gplane, blk, 0, stream>>>(parent, count);
  k_maxarea<<<gplane, blk, 0, stream>>>(count, scal);
  k_keep<<<gimg, blk, 0, stream>>>(parent, count, scal, cams, keepfg, keepbg, scal + 16, scal + 17);
  k_ce<<<dim3(GX, 1, 3), blk, 0, stream>>>(preds, keepfg, keepbg, part);
  k_final<<<1, 64, 0, stream>>>(part, scal, (float*)d_out);
}